// GCNConv_34007551050420
// MI455X (gfx1250) — compile-verified
//
#include <hip/hip_runtime.h>
#include <hip/hip_bf16.h>

// GCNConv: out = D^{-1/2} (A + I) D^{-1/2} (x @ W) + b
// N=50000 (=3125*16), E=800000, D_IN=D_OUT=96 (=6*16=24*4)

#define DFEAT 96

typedef __attribute__((ext_vector_type(2))) float v2f;
typedef __attribute__((ext_vector_type(8))) float v8f;

// ---------------- degree / normalization ----------------

__global__ void deg_init(float* __restrict__ deg, int N) {
  int i = blockIdx.x * blockDim.x + threadIdx.x;
  if (i < N) deg[i] = 1.0f;  // self-loop weight 1
}

__global__ void deg_accum(const long long* __restrict__ ei, const float* __restrict__ ew,
                          float* __restrict__ deg, int E) {
  int e = blockIdx.x * blockDim.x + threadIdx.x;
  if (e < E) {
    int c = (int)ei[(size_t)E + e];  // col = edge_index[1][e]
    atomicAdd(&deg[c], ew[e]);
  }
}

__global__ void deg_rsqrt(float* __restrict__ deg, int N) {
  int i = blockIdx.x * blockDim.x + threadIdx.x;
  if (i < N) {
    float d = deg[i];
    deg[i] = (d > 0.0f) ? rsqrtf(d) : 0.0f;  // deg -> dinv in place
  }
}

// ---------------- h = x @ W via V_WMMA_F32_16X16X4_F32 ----------------
// One wave (32 lanes) per 16x16 output tile; grid = (96/16, N/16).
// A 16x4 f32 fragment (2 VGPRs): lanes 0-15 hold row M=lane, K={k,k+1};
//   lanes 16-31 hold row M=lane-16, K={k+2,k+3}.
// B 4x16 f32 fragment (2 VGPRs, mirrored): lanes 0-15 hold col N=lane,
//   K={k,k+1}; lanes 16-31 hold col N=lane-16, K={k+2,k+3}.
// C/D 16x16 f32 (8 VGPRs): VGPR r = row r (lanes 0-15) / row r+8 (lanes 16-31).
__global__ void gemm_wmma(const float* __restrict__ x, const float* __restrict__ W,
                          float* __restrict__ h) {
  const int lane = threadIdx.x;        // blockDim.x == 32, full wave
  const int half = lane >> 4;          // 0 or 1
  const int l    = lane & 15;
  const int n0   = blockIdx.x * 16;    // output column tile
  const int m0   = blockIdx.y * 16;    // output row tile

  const float* arow = x + (size_t)(m0 + l) * DFEAT + 2 * half;
  const float* bcol = W + (size_t)(2 * half) * DFEAT + (n0 + l);

  v8f acc = {};
#pragma unroll
  for (int k = 0; k < DFEAT; k += 4) {
    v2f a, b;
    a.x = arow[k];
    a.y = arow[k + 1];
    b.x = bcol[(size_t)k * DFEAT];
    b.y = bcol[(size_t)(k + 1) * DFEAT];
    acc = __builtin_amdgcn_wmma_f32_16x16x4_f32(
        /*neg_a=*/false, a, /*neg_b=*/false, b,
        /*c_mod=*/(short)0, acc, /*reuse_a=*/false, /*reuse_b=*/false);
  }

  float* o = h + (size_t)(m0 + half * 8) * DFEAT + n0 + l;
#pragma unroll
  for (int r = 0; r < 8; ++r) o[(size_t)r * DFEAT] = acc[r];
}

// Scalar fallback for a row remainder (not launched when N % 16 == 0).
__global__ void gemm_tail(const float* __restrict__ x, const float* __restrict__ W,
                          float* __restrict__ h, int rowStart, int N) {
  int t = blockIdx.x * blockDim.x + threadIdx.x;
  int row = rowStart + t / DFEAT;
  int col = t % DFEAT;
  if (row < N) {
    float s = 0.0f;
    for (int k = 0; k < DFEAT; ++k) s += x[(size_t)row * DFEAT + k] * W[(size_t)k * DFEAT + col];
    h[(size_t)row * DFEAT + col] = s;
  }
}

// ---------------- out init: self-loop term + bias ----------------
// out[i,:] = h[i,:] * dinv[i]^2 + b ; 24 threads per node, float4 each.
__global__ void init_out(const float* __restrict__ h, const float* __restrict__ dinv,
                         const float* __restrict__ b, float* __restrict__ out, int N) {
  int t = blockIdx.x * blockDim.x + threadIdx.x;
  int i = t / 24;
  int d4 = (t % 24) * 4;
  if (i < N) {
    float di = dinv[i];
    float s = di * di;
    const float4 hv = *(const float4*)(h + (size_t)i * DFEAT + d4);
    const float4 bv = *(const float4*)(b + d4);
    float4 ov;
    ov.x = hv.x * s + bv.x;
    ov.y = hv.y * s + bv.y;
    ov.z = hv.z * s + bv.z;
    ov.w = hv.w * s + bv.w;
    *(float4*)(out + (size_t)i * DFEAT + d4) = ov;
  }
}

// ---------------- edge scatter (dominant phase) ----------------
// 24 threads per edge; each gathers float4 of h[row] and atomically adds
// norm-scaled values into out[col]. 384 contiguous bytes per edge group.
__global__ void edge_scatter(const long long* __restrict__ ei, const float* __restrict__ ew,
                             const float* __restrict__ dinv, const float* __restrict__ h,
                             float* __restrict__ out, int E) {
  long long t = (long long)blockIdx.x * blockDim.x + threadIdx.x;
  int e = (int)(t / 24);
  int d4 = (int)(t % 24) * 4;
  if (e < E) {
    int r = (int)ei[e];               // row = source
    int c = (int)ei[(size_t)E + e];   // col = target
    float norm = dinv[r] * ew[e] * dinv[c];
    const float4 hv = *(const float4*)(h + (size_t)r * DFEAT + d4);
    float* o = out + (size_t)c * DFEAT + d4;
    atomicAdd(o + 0, hv.x * norm);
    atomicAdd(o + 1, hv.y * norm);
    atomicAdd(o + 2, hv.z * norm);
    atomicAdd(o + 3, hv.w * norm);
  }
}

extern "C" void kernel_launch(void* const* d_in, const int* in_sizes, int n_in,
                              void* d_out, int out_size, void* d_ws, size_t ws_size,
                              hipStream_t stream) {
  const float*     x  = (const float*)d_in[0];      // [N, 96]
  const long long* ei = (const long long*)d_in[1];  // [2, E] int64
  const float*     ew = (const float*)d_in[2];      // [E]
  const float*     W  = (const float*)d_in[3];      // [96, 96]
  const float*     b  = (const float*)d_in[4];      // [96]
  float*           out = (float*)d_out;             // [N, 96]

  const int N = in_sizes[0] / DFEAT;
  const int E = in_sizes[2];

  // Workspace: dinv (N floats) then h (N*96 floats), 256B aligned.
  float* dinv = (float*)d_ws;
  size_t hoff = ((size_t)N * sizeof(float) + 255) & ~(size_t)255;
  float* h = (float*)((char*)d_ws + hoff);

  // 1) degree with self-loops
  deg_init<<<(N + 255) / 256, 256, 0, stream>>>(dinv, N);
  deg_accum<<<(E + 255) / 256, 256, 0, stream>>>(ei, ew, dinv, E);
  deg_rsqrt<<<(N + 255) / 256, 256, 0, stream>>>(dinv, N);

  // 2) h = x @ W  (WMMA f32 16x16x4; exact tiling for N=50000, D=96)
  int mblocks = N / 16;
  if (mblocks > 0) {
    dim3 grid(DFEAT / 16, mblocks);
    gemm_wmma<<<grid, 32, 0, stream>>>(x, W, h);
  }
  int rem = N - mblocks * 16;
  if (rem > 0) {
    int tailThreads = rem * DFEAT;
    gemm_tail<<<(tailThreads + 255) / 256, 256, 0, stream>>>(x, W, h, mblocks * 16, N);
  }

  // 3) out = h * dinv^2 + b  (self-loop term)
  {
    long long t = (long long)N * 24;
    init_out<<<(unsigned)((t + 255) / 256), 256, 0, stream>>>(h, dinv, b, out, N);
  }

  // 4) scatter-add over edges
  {
    long long t = (long long)E * 24;
    edge_scatter<<<(unsigned)((t + 255) / 256), 256, 0, stream>>>(ei, ew, dinv, h, out, E);
  }
}